// e3LayerNorm_9972914061338
// MI455X (gfx1250) — compile-verified
//
#include <hip/hip_runtime.h>
#include <math.h>

// ---------------- problem constants (from reference) ----------------
#define N_NODES   200000
#define N_GRAPHS  64
#define DIM_TOT   240          // 64*1 + 32*3 + 16*5
#define QPR       60           // float4 per row
#define TILES     (N_NODES / 16)
#define NSTAT     16           // padded per-graph stat stride (11 used)
#define EPSV      1e-5f

typedef __attribute__((ext_vector_type(2))) float v2f;
typedef __attribute__((ext_vector_type(4))) float v4f;
typedef __attribute__((ext_vector_type(8))) float v8f;

// ---------------- pass 0: zero the stats accumulators ----------------
__global__ void e3ln_zero(float* stats) {
  int i = blockIdx.x * blockDim.x + threadIdx.x;
  if (i < N_GRAPHS * NSTAT) stats[i] = 0.0f;
}

// ---------------- pass 1: per-graph stats via WMMA reduction ----------------
// Each wave owns a 16-node tile. The 240->10 row reduction is
//   C(16x16) += A(16x4 of x) * S(4x16)   over 60 K-chunks  (V_WMMA_F32_16X16X4_F32)
// plus a second WMMA with A = x*x and a column-1 selector over chunks 0..15
// (channels 0..63) for the sum of squares.
// Selection-matrix columns: c<64 -> 0 (x) / 1 (x^2); l=1 -> 2+(c-64)%3;
// l=2 -> 5+(c-160)%5.  B is constant over the scalar block, period-3 over the
// l=1 block and period-5 over the l=2 block, so all B operands are hoisted
// into registers: the steady-state inner loop is load_b64 + wmma only.
__global__ void __launch_bounds__(256)
e3ln_stats(const float* __restrict__ x, const int* __restrict__ batch,
           float* __restrict__ stats) {
  __shared__ float sacc[N_GRAPHS * 12];
  for (int i = threadIdx.x; i < N_GRAPHS * 12; i += blockDim.x) sacc[i] = 0.0f;
  __syncthreads();

  const int lane  = threadIdx.x & 31;
  const int ncol  = lane & 15;        // stat column this lane holds in C/D
  const int half  = lane >> 4;        // K-half for the A/B operand layout
  const int koff  = half * 2;         // this lane supplies K = koff, koff+1
  const int wave  = (blockIdx.x * blockDim.x + threadIdx.x) >> 5;
  const int nwave = (gridDim.x * blockDim.x) >> 5;

  // Hoisted B operands (computed once per lane).
  const float s0 = (ncol == 0) ? 1.0f : 0.0f;   // x   -> column 0 (c < 64)
  const float s1 = (ncol == 1) ? 1.0f : 0.0f;   // x^2 -> column 1 (c < 64)
  v2f b0; b0.x = s0; b0.y = s0;
  v2f bsq; bsq.x = s1; bsq.y = s1;
  v2f bL1[3], bL2[5];
#pragma unroll
  for (int p = 0; p < 3; ++p) {                 // l=1: col 2 + (c-64)%3
    const int cc = 4 * p + koff;
    bL1[p].x = ((2 + (cc % 3))       == ncol) ? 1.0f : 0.0f;
    bL1[p].y = ((2 + ((cc + 1) % 3)) == ncol) ? 1.0f : 0.0f;
  }
#pragma unroll
  for (int p = 0; p < 5; ++p) {                 // l=2: col 5 + (c-160)%5
    const int cc = 4 * p + koff;
    bL2[p].x = ((5 + (cc % 5))       == ncol) ? 1.0f : 0.0f;
    bL2[p].y = ((5 + ((cc + 1) % 5)) == ncol) ? 1.0f : 0.0f;
  }

  for (int tile = wave; tile < TILES; tile += nwave) {
    const int   rowL = tile * 16 + ncol;                 // lanes 16..31 mirror rows 0..15
    const int   g    = batch[rowL];
    const float* rp  = x + (size_t)rowL * DIM_TOT + koff;

    // prefetch next tile's rows (global_prefetch_b8)
    if (tile + nwave < TILES) {
      const char* nx = (const char*)(x + (size_t)(tile + nwave) * 16 * DIM_TOT);
      __builtin_prefetch(nx + lane * 480, 0, 1);
    }

    v8f C = {};
    // chunks 0..15: channels 0..63 -> scalar sum (col 0) + sum of squares (col 1)
#pragma unroll 4
    for (int kc = 0; kc < 16; ++kc) {
      v2f a = *(const v2f*)(rp + kc * 4);
      C = __builtin_amdgcn_wmma_f32_16x16x4_f32(false, a, false, b0,
                                                (short)0, C, false, false);
      v2f a2; a2.x = a.x * a.x; a2.y = a.y * a.y;
      C = __builtin_amdgcn_wmma_f32_16x16x4_f32(false, a2, false, bsq,
                                                (short)0, C, false, false);
    }
    // chunks 16..39: l=1 component sums (B period 3)
    for (int base = 16; base < 40; base += 3) {
#pragma unroll
      for (int p = 0; p < 3; ++p) {
        v2f a = *(const v2f*)(rp + (base + p) * 4);
        C = __builtin_amdgcn_wmma_f32_16x16x4_f32(false, a, false, bL1[p],
                                                  (short)0, C, false, false);
      }
    }
    // chunks 40..59: l=2 component sums (B period 5)
    for (int base = 40; base < 60; base += 5) {
#pragma unroll
      for (int p = 0; p < 5; ++p) {
        v2f a = *(const v2f*)(rp + (base + p) * 4);
        C = __builtin_amdgcn_wmma_f32_16x16x4_f32(false, a, false, bL2[p],
                                                  (short)0, C, false, false);
      }
    }

    // C layout: VGPR r, lanes 0-15 -> (M=r, N=lane); lanes 16-31 -> (M=r+8, N=lane-16)
#pragma unroll
    for (int r = 0; r < 8; ++r) {
      const int m  = r + (half << 3);
      const int gm = __shfl(g, m, 32);                   // graph of node row m
      const float v = C[r];
      if (ncol < 10)        atomicAdd(&sacc[gm * 12 + ncol], v);
      else if (ncol == 10)  atomicAdd(&sacc[gm * 12 + 10], 1.0f);  // node count
    }
  }

  __syncthreads();
  for (int i = threadIdx.x; i < N_GRAPHS * 11; i += blockDim.x) {
    const int g = i / 11, t = i - g * 11;
    atomicAdd(&stats[g * NSTAT + t], sacc[g * 12 + t]);
  }
}

// ---------------- pass 2: fold stats into per-(graph,channel) A/B tables ----
__global__ void e3ln_tables(const float* __restrict__ stats,
                            const float* __restrict__ weight,
                            const float* __restrict__ bias,
                            float* __restrict__ tabA, float* __restrict__ tabB) {
  int i = blockIdx.x * blockDim.x + threadIdx.x;
  if (i >= N_GRAPHS * DIM_TOT) return;
  const int g = i / DIM_TOT, c = i - g * DIM_TOT;
  const float* s = stats + g * NSTAT;
  const float cnt = s[10] < 1.0f ? 1.0f : s[10];
  float A, B;
  if (c < 64) {                         // scalar irrep: center, normalize, affine
    const float inv  = 1.0f / (cnt * 64.0f);
    const float ms   = s[0] * inv;                       // grand mean
    float nrm        = s[1] * inv - ms * ms;             // E[x^2]-m^2 = E[(x-m)^2]
    nrm              = nrm > 0.0f ? nrm : 0.0f;
    const float rdiv = 1.0f / (sqrtf(nrm) + EPSV);
    const float w    = weight[c];
    A = rdiv * w;  B = bias[c] - ms * rdiv * w;
  } else if (c < 160) {                 // l=1: center per component, scale
    const int cc = c - 64, j = cc / 3, d = cc - j * 3;
    const float mean = s[2 + d] / (cnt * 32.0f);
    const float w    = weight[64 + j];
    A = w;  B = -mean * w;
  } else {                              // l=2
    const int cc = c - 160, j = cc / 5, d = cc - j * 5;
    const float mean = s[5 + d] / (cnt * 16.0f);
    const float w    = weight[96 + j];
    A = w;  B = -mean * w;
  }
  tabA[i] = A;  tabB[i] = B;
}

// ---------------- pass 3: streaming apply, NT stores --------------------
// out[n,c] = x[n,c]*A[batch[n],c] + B[batch[n],c].  x re-read hits L2 (192MB),
// output leaves via non-temporal stores so it does not evict x.
__global__ void __launch_bounds__(256)
e3ln_apply(const float* __restrict__ x, const int* __restrict__ batch,
           const float* __restrict__ tabA, const float* __restrict__ tabB,
           float* __restrict__ out) {
  const int NQ = N_NODES * QPR;        // 12,000,000 float4s
  int i = blockIdx.x * blockDim.x + threadIdx.x;
  const int stride = gridDim.x * blockDim.x;
  const v4f* __restrict__ xv4 = (const v4f*)x;
  const v4f* __restrict__ av4 = (const v4f*)tabA;
  const v4f* __restrict__ bv4 = (const v4f*)tabB;
  v4f* __restrict__ ov4 = (v4f*)out;
  for (; i < NQ; i += stride) {
    const int node = i / QPR;
    const int q    = i - node * QPR;
    const int g    = batch[node];
    const v4f xv = xv4[i];
    const v4f av = av4[g * QPR + q];
    const v4f bv = bv4[g * QPR + q];
    v4f o;
    o.x = fmaf(xv.x, av.x, bv.x);
    o.y = fmaf(xv.y, av.y, bv.y);
    o.z = fmaf(xv.z, av.z, bv.z);
    o.w = fmaf(xv.w, av.w, bv.w);
    __builtin_nontemporal_store(o, &ov4[i]);
  }
}

// ---------------- launcher ----------------
extern "C" void kernel_launch(void* const* d_in, const int* in_sizes, int n_in,
                              void* d_out, int out_size, void* d_ws, size_t ws_size,
                              hipStream_t stream) {
  const float* x      = (const float*)d_in[0];
  const int*   batch  = (const int*)d_in[1];
  const float* weight = (const float*)d_in[2];
  const float* bias   = (const float*)d_in[3];
  float*       out    = (float*)d_out;

  float* stats = (float*)d_ws;                       // 64*16 floats
  float* tabA  = stats + 1024;                       // 64*240 floats (16B aligned)
  float* tabB  = tabA + N_GRAPHS * DIM_TOT;          // 64*240 floats

  e3ln_zero<<<1, 1024, 0, stream>>>(stats);
  e3ln_stats<<<256, 256, 0, stream>>>(x, batch, stats);
  e3ln_tables<<<(N_GRAPHS * DIM_TOT + 255) / 256, 256, 0, stream>>>(
      stats, weight, bias, tabA, tabB);
  e3ln_apply<<<4096, 256, 0, stream>>>(x, batch, tabA, tabB, out);
}